// OutputNeuron_4604204941431
// MI455X (gfx1250) — compile-verified
//
#include <hip/hip_runtime.h>
#include <stdint.h>

// SNN scan: syn = A*syn + x; mem = B*mem + syn; s = mem>1; count += s; mem *= (1-s)
// Memory-bound streaming kernel (~274 MB @ 23.3 TB/s => ~12 us floor).
// gfx1250 async global->LDS loads (ASYNCcnt) used as a per-thread prefetch FIFO:
// 16 timesteps x 16 B in flight per thread with zero VGPR cost.

#define SNN_ALPHA 0.9f
#define SNN_BETA  0.95f

constexpr int BLOCK = 256;   // 8 wave32 per block
constexpr int DEPTH = 16;    // async prefetch depth (timesteps in flight)

__global__ __launch_bounds__(BLOCK)
void snn_scan_kernel(const float* __restrict__ x,
                     const float* __restrict__ mem0,
                     const float* __restrict__ syn0,
                     const float* __restrict__ count0,
                     float* __restrict__ out,
                     int T, int BN)
{
    // Per-thread 16B slots, DEPTH deep: 16 * 256 * 16B = 64 KB LDS per block.
    __shared__ float4 stage[DEPTH][BLOCK];

    const int tid  = threadIdx.x;
    const int elt4 = blockIdx.x * BLOCK + tid;        // float4 index into [B*N]
    const uint32_t thr_off = (uint32_t)elt4 * 16u;    // byte offset within one timestep
    const uint32_t tstride = (uint32_t)BN * 4u;       // bytes between timesteps (2 MB)
    const uint64_t sbase   = (uint64_t)(uintptr_t)x;  // uniform SGPR base (GVS mode)
    const uint32_t lds0    = (uint32_t)(uintptr_t)&stage[0][tid];
    constexpr uint32_t SLOT_STRIDE = (uint32_t)(BLOCK * sizeof(float4)); // 4 KB

    // ---- Prologue: issue min(DEPTH,T) async loads (runtime loop, runs once) ----
    {
        const int npro = (T < DEPTH) ? T : DEPTH;
        uint32_t voff = thr_off;
        uint32_t lds  = lds0;
#pragma unroll 1
        for (int t = 0; t < npro; ++t) {
            asm volatile("global_load_async_to_lds_b128 %0, %1, %2"
                         :: "v"(lds), "v"(voff), "s"(sbase) : "memory");
            voff += tstride;
            lds  += SLOT_STRIDE;
        }
    }

    // Initial state (zeros in the harness, but read them for generality).
    float4 mem = ((const float4*)mem0)[elt4];
    float4 syn = ((const float4*)syn0)[elt4];
    float4 cnt = ((const float4*)count0)[elt4];

#define SNN_STEP(v, c)                                      \
        {                                                   \
            syn.c = __builtin_fmaf(SNN_ALPHA, syn.c, v.c);  \
            mem.c = __builtin_fmaf(SNN_BETA,  mem.c, syn.c);\
            bool fired = (mem.c > 1.0f);                    \
            cnt.c += fired ? 1.0f : 0.0f;                   \
            mem.c  = fired ? 0.0f : mem.c;                  \
        }

    // ---- Steady state: t in [0, T-DEPTH). Load t is the oldest of DEPTH
    // outstanding; ASYNCcnt completes in order, so asynccnt <= DEPTH-1 means
    // slot t has landed. Reissue the slot for t+DEPTH after draining the read.
    int t = 0;
    {
        uint32_t voff_next = thr_off + (uint32_t)DEPTH * tstride; // byte addr of load t+DEPTH
#pragma unroll 1
        for (; t < T - DEPTH; ++t) {
            const int slot = t & (DEPTH - 1);

            asm volatile("s_wait_asynccnt 15" ::: "memory");
            float4 xv = stage[slot][tid];                 // ds_load_b128
            asm volatile("s_wait_dscnt 0" ::: "memory");  // read done before slot reuse

            uint32_t lds = lds0 + (uint32_t)slot * SLOT_STRIDE;
            asm volatile("global_load_async_to_lds_b128 %0, %1, %2"
                         :: "v"(lds), "v"(voff_next), "s"(sbase) : "memory");
            voff_next += tstride;

            SNN_STEP(xv, x) SNN_STEP(xv, y) SNN_STEP(xv, z) SNN_STEP(xv, w)
        }
    }

    // ---- Drain: all remaining timesteps already issued; wait once, then
    // consume from LDS with no further waits.
    asm volatile("s_wait_asynccnt 0" ::: "memory");
#pragma unroll 1
    for (; t < T; ++t) {
        const int slot = t & (DEPTH - 1);
        float4 xv = stage[slot][tid];
        SNN_STEP(xv, x) SNN_STEP(xv, y) SNN_STEP(xv, z) SNN_STEP(xv, w)
    }
#undef SNN_STEP

    ((float4*)out)[elt4] = cnt;   // final spike counts, [B,N] float32
}

extern "C" void kernel_launch(void* const* d_in, const int* in_sizes, int n_in,
                              void* d_out, int out_size, void* d_ws, size_t ws_size,
                              hipStream_t stream)
{
    const float* x      = (const float*)d_in[0];
    const float* mem0   = (const float*)d_in[1];
    const float* syn0   = (const float*)d_in[2];
    const float* count0 = (const float*)d_in[3];
    float*       out    = (float*)d_out;

    const int BN = in_sizes[1];            // B*N (mem0 flat size) = 524288
    const int T  = in_sizes[0] / BN;       // 128

    // BN = 524288 is divisible by BLOCK*4 = 1024 -> exact grid, no tail.
    const int blocks = BN / (BLOCK * 4);   // 512

    snn_scan_kernel<<<blocks, BLOCK, 0, stream>>>(x, mem0, syn0, count0, out, T, BN);
}